// Walk_51780125721453
// MI455X (gfx1250) — compile-verified
//
#include <hip/hip_runtime.h>

typedef float v2f __attribute__((ext_vector_type(2)));
typedef float v8f __attribute__((ext_vector_type(8)));

#define BN   16
#define C    64
#define TOT  8192
#define KNB  20
#define CN   512
#define STEPS 8
#define WPB  8          // waves (walkers) per block
#define BN_EPS 1e-5f

// ---------------- wave32 reductions (butterfly: result in all lanes) ----------------
__device__ __forceinline__ float wsumf(float v) {
#pragma unroll
  for (int m = 16; m >= 1; m >>= 1) v += __shfl_xor(v, m, 32);
  return v;
}
__device__ __forceinline__ float wmaxf(float v) {
#pragma unroll
  for (int m = 16; m >= 1; m >>= 1) v = fmaxf(v, __shfl_xor(v, m, 32));
  return v;
}
__device__ __forceinline__ int wmini(int v) {
#pragma unroll
  for (int m = 16; m >= 1; m >>= 1) { int o = __shfl_xor(v, m, 32); v = (o < v) ? o : v; }
  return v;
}

// ---------------- transpose x: (bn, c, tot) -> xt: (bn, tot, c) ----------------
// Makes every gathered feature row a contiguous 256B line (table is 33.5MB -> L2-resident).
__global__ __launch_bounds__(256) void xpose_kernel(const float* __restrict__ x,
                                                    float* __restrict__ xt) {
  __shared__ float tile[32][33];
  const int b  = blockIdx.z;
  const int c0 = blockIdx.y * 32;
  const int t0 = blockIdx.x * 32;
  const int tx = threadIdx.x, ty = threadIdx.y;
#pragma unroll
  for (int i = ty; i < 32; i += 8)
    tile[i][tx] = x[((size_t)(b * C + c0 + i)) * TOT + t0 + tx];
  __syncthreads();
#pragma unroll
  for (int i = ty; i < 32; i += 8)
    xt[((size_t)(b * TOT + t0 + i)) * C + c0 + tx] = tile[tx][i];
}

// ---------------- main walker kernel: one wave32 per walker ----------------
__global__ __launch_bounds__(256) void walk_kernel(
    const float* __restrict__ xt, const int* __restrict__ adj, const int* __restrict__ cur,
    const float* __restrict__ agent_w,
    const float* __restrict__ ag, const float* __restrict__ ab,
    const float* __restrict__ am, const float* __restrict__ av,
    const float* __restrict__ mom_w,
    const float* __restrict__ mg, const float* __restrict__ mb,
    const float* __restrict__ mm, const float* __restrict__ mv,
    float* __restrict__ out) {
  // Per-wave query matrix Q: rows = {agent_w_lo, cur-pre, cur, ZERO}.
  // Row 3 stays all-zero so lanes q>=3 feed zeros into WMMA A without any VALU mask.
  __shared__ float Q[WPB][4][C];

  const int lane   = threadIdx.x & 31;
  const int warp   = threadIdx.x >> 5;
  const int walker = blockIdx.x * WPB + warp;
  const int b      = walker / CN;
  const int n      = walker % CN;

  // BN folded affines (uniform scalars)
  const float invA  = ag[0] / sqrtf(av[0] + BN_EPS);
  const float aMean = am[0], aBeta = ab[0];
  const float invM0 = mg[0] / sqrtf(mv[0] + BN_EPS);
  const float invM1 = mg[1] / sqrtf(mv[1] + BN_EPS);
  const float mM0 = mm[0], mM1 = mm[1], mB0 = mb[0], mB1 = mb[1];

  // Each lane owns channel pair cc, cc+1
  const int cc = lane * 2;
  const float whi0  = agent_w[C + cc],       whi1  = agent_w[C + cc + 1];
  const float mw0c0 = mom_w[cc],             mw0c1 = mom_w[cc + 1];
  const float mw0p0 = mom_w[C + cc],         mw0p1 = mom_w[C + cc + 1];
  const float mw1c0 = mom_w[2 * C + cc],     mw1c1 = mom_w[2 * C + cc + 1];
  const float mw1p0 = mom_w[3 * C + cc],     mw1p1 = mom_w[3 * C + cc + 1];

  Q[warp][0][cc] = agent_w[cc]; Q[warp][0][cc + 1] = agent_w[cc + 1];
  Q[warp][1][cc] = 0.f;         Q[warp][1][cc + 1] = 0.f;
  Q[warp][2][cc] = 0.f;         Q[warp][2][cc + 1] = 0.f;
  Q[warp][3][cc] = 0.f;         Q[warp][3][cc + 1] = 0.f;   // permanent zero row

  // walker state
  int node = cur[b * CN + n];
  const float* r = xt + ((size_t)(b * TOT + node)) * C;
  float fp0 = r[cc], fp1 = r[cc + 1];          // pre_feature (register-resident)
  float fc0 = 0.f,  fc1 = 0.f;                 // cur_feature
  float dwp = wsumf(whi0 * fp0 + whi1 * fp1);  // agent_w_hi . pre_feature

  // WMMA fragment lane roles (16x16x4 f32): lanes 0-15 carry K=0,1; 16-31 carry K=2,3
  const int q    = lane & 15;   // A-row / B-column index within a tile
  const int coff = (lane >> 4) * 2;
  const float* qp = &Q[warp][(q < 3) ? q : 3][coff];  // q>=3 -> zero row (no mask VALU)

  for (int s = 0; s < STEPS; ++s) {
    if (s > 0) {
      // momentum attention: mlog = mom_w @ [cur; pre], BN, softmax2
      float m0 = wsumf(mw0c0 * fc0 + mw0c1 * fc1 + mw0p0 * fp0 + mw0p1 * fp1);
      float m1 = wsumf(mw1c0 * fc0 + mw1c1 * fc1 + mw1p0 * fp0 + mw1p1 * fp1);
      m0 = (m0 - mM0) * invM0 + mB0;
      m1 = (m1 - mM1) * invM1 + mB1;
      float mx2 = fmaxf(m0, m1);
      float e0 = __expf(m0 - mx2), e1 = __expf(m1 - mx2);
      float inv = 1.f / (e0 + e1);
      float att0 = e0 * inv, att1 = e1 * inv;
      fp0 = att0 * fc0 + att1 * fp0;
      fp1 = att0 * fc1 + att1 * fp1;
      dwp = wsumf(whi0 * fp0 + whi1 * fp1);
      Q[warp][1][cc] = fc0 - fp0; Q[warp][1][cc + 1] = fc1 - fp1;
      Q[warp][2][cc] = fc0;       Q[warp][2][cc + 1] = fc1;
    }
    __syncthreads();

    // neighbor indices (tile0: k=0..15 on column q; tile1: k=16..19, clamped)
    const int* arow = adj + ((size_t)(b * TOT + node)) * KNB;
    const int idx0 = arow[q];
    int k1 = 16 + q; k1 = (k1 > KNB - 1) ? (KNB - 1) : k1;
    const int idx1 = arow[k1];
    const float* r0 = xt + ((size_t)(b * TOT + idx0)) * C + coff;
    const float* r1 = xt + ((size_t)(b * TOT + idx1)) * C + coff;

    // D[q, nb] = sum_c Q[q][c] * X[nb][c] via chained f32 WMMA over 16 K-chunks
    v8f acc0 = {}; v8f acc1 = {};
    float ss0 = 0.f, ss1 = 0.f;       // per-half sum of squares (self-dots)
#pragma unroll
    for (int ch = 0; ch < 16; ++ch) {
      v2f a  = *(const v2f*)(qp + ch * 4);
      v2f b0 = *(const v2f*)(r0 + ch * 4);
      v2f b1 = *(const v2f*)(r1 + ch * 4);
      acc0 = __builtin_amdgcn_wmma_f32_16x16x4_f32(false, a, false, b0, (short)0, acc0, false, false);
      acc1 = __builtin_amdgcn_wmma_f32_16x16x4_f32(false, a, false, b1, (short)0, acc1, false, false);
      ss0 += b0.x * b0.x + b0.y * b0.y;
      ss1 += b1.x * b1.x + b1.y * b1.y;
    }

    // combine self-dot halves: lanes j and j+16 both end with neighbor j's total
    float selfA = ss0 + __shfl_xor(ss0, 16, 32);
    float selfB = ss1 + __shfl_xor(ss1, 16, 32);
    // tile0 dots land lane-aligned (lane n = neighbor n) in acc rows 0..2;
    // tile1 dots (lanes 0..3) are shuffled to lanes 16..19
    float dwB = __shfl(acc1[0], q, 32);
    float dvB = __shfl(acc1[1], q, 32);
    float dcB = __shfl(acc1[2], q, 32);
    const bool hi = (lane >= 16);
    float dw  = hi ? dwB   : acc0[0];   // agent_w_lo . P_k
    float dv  = hi ? dvB   : acc0[1];   // (cur-pre)  . P_k
    float dc  = hi ? dcB   : acc0[2];   // cur        . P_k
    float slf = hi ? selfB : selfA;     // P_k . P_k
    int myidx = hi ? idx1  : idx0;
    const bool valid = (lane < KNB);

    // logits + BN affine
    float lg = ((dw + dwp) - aMean) * invA + aBeta;
    if (s > 0) {
      // crossover gate: cos(cur-pre, P_k - cur), d = clamp(1+cos, 0, 1)
      float v0 = fc0 - fp0, v1 = fc1 - fp1;
      float n1sq = wsumf(v0 * v0 + v1 * v1);
      float vdfc = wsumf(v0 * fc0 + v1 * fc1);
      float fc2  = wsumf(fc0 * fc0 + fc1 * fc1);
      float n2   = sqrtf(fmaxf(slf - 2.f * dc + fc2, 0.f));
      float cosv = (dv - vdfc) / fmaxf(sqrtf(n1sq) * n2, 1e-8f);
      lg *= fminf(fmaxf(1.f + cosv, 0.f), 1.f);
    }
    if (!valid) lg = -__builtin_inff();

    // straight-through gumbel: one-hot(argmax) with (1-y)+y scaling artifact
    float mx  = wmaxf(lg);
    int cand  = (lg == mx) ? lane : 64;       // first-max tie-break like jnp.argmax
    int amax  = wmini(cand);
    float sum = wsumf(valid ? __expf(lg - mx) : 0.f);
    float yam = 1.f / sum;
    float scale = (1.f - yam) + yam;

    node = __shfl(myidx, amax, 32);
    const float* rs = xt + ((size_t)(b * TOT + node)) * C;
    fc0 = rs[cc] * scale;
    fc1 = rs[cc + 1] * scale;

    // out layout (bn, c, cn, STEPS)
    out[(((size_t)b * C + cc) * CN + n) * STEPS + s]     = fc0;
    out[(((size_t)b * C + cc + 1) * CN + n) * STEPS + s] = fc1;
    __syncthreads();
  }
}

extern "C" void kernel_launch(void* const* d_in, const int* in_sizes, int n_in,
                              void* d_out, int out_size, void* d_ws, size_t ws_size,
                              hipStream_t stream) {
  // inputs: 0 xyz (unused), 1 x, 2 adj, 3 cur, 4 agent_w, 5..8 agent bn, 9 mom_w, 10..13 mom bn
  const float* x       = (const float*)d_in[1];
  const int*   adj     = (const int*)d_in[2];
  const int*   cur     = (const int*)d_in[3];
  const float* agent_w = (const float*)d_in[4];
  const float* ag      = (const float*)d_in[5];
  const float* ab      = (const float*)d_in[6];
  const float* am      = (const float*)d_in[7];
  const float* av      = (const float*)d_in[8];
  const float* mom_w   = (const float*)d_in[9];
  const float* mg      = (const float*)d_in[10];
  const float* mb      = (const float*)d_in[11];
  const float* mm      = (const float*)d_in[12];
  const float* mv      = (const float*)d_in[13];

  float* xt = (float*)d_ws;  // 16*8192*64*4 = 32 MB transposed feature table

  dim3 gT(TOT / 32, C / 32, BN), bT(32, 8);
  xpose_kernel<<<gT, bT, 0, stream>>>(x, xt);

  walk_kernel<<<(BN * CN) / WPB, 32 * WPB, 0, stream>>>(
      xt, adj, cur, agent_w, ag, ab, am, av, mom_w, mg, mb, mm, mv, (float*)d_out);
}